// Conv_62268435857617
// MI455X (gfx1250) — compile-verified
//
#include <hip/hip_runtime.h>
#include <stdint.h>

typedef float v2f __attribute__((ext_vector_type(2)));
typedef float v8f __attribute__((ext_vector_type(8)));
typedef unsigned int v4u __attribute__((ext_vector_type(4)));
typedef int v4i __attribute__((ext_vector_type(4)));
typedef int v8i __attribute__((ext_vector_type(8)));

#define IMG_FLOATS   128                    // 16x8 image = 128 f32
#define IMGS_PER_WG  128                    // 8 waves * 16 images
#define N_TILES      60                     // nonzero (ptile,kstep) pairs of Kexp
#define KEXP_FLOATS  (N_TILES * 64)         // 64 f32 per WMMA A-tile (2 VGPR * 32 lanes)
#define KEXP_BYTES   (KEXP_FLOATS * 4)      // 15360 B
#define IMG_REGION_B (IMGS_PER_WG * IMG_FLOATS * 4)  // 65536 B
#define SMEM_BYTES   (KEXP_BYTES + IMG_REGION_B)     // 80896 B (<320KB/WGP)

__global__ __launch_bounds__(256, 1) void conv_relu_wmma(
    const float* __restrict__ x, const float* __restrict__ kern,
    float* __restrict__ out)
{
  extern __shared__ char smem[];
  float* kexp = (float*)smem;                       // Kexp tiles, WMMA A layout
  float* imgs = (float*)(smem + KEXP_BYTES);        // staged image data

  const int tid  = threadIdx.x;
  const int lane = tid & 31;
  const int wave = tid >> 5;
  const int h    = lane >> 4;      // lane half (0/1)
  const int ln   = lane & 15;

  // banded structure of Kexp: per output-row tile t, active kstep range
  const int kslo[6]  = {0, 4, 10, 16, 20, 26};
  const int kshi[6]  = {10, 16, 20, 26, 32, 32};
  const int tbase[6] = {0, 10, 22, 32, 42, 54};

  // ---------- stage this wave's 16 images (8 KB) into LDS via TDM ----------
  const size_t imgBlock = (size_t)blockIdx.x * IMGS_PER_WG;
  const float* gsrc = x + (imgBlock + (size_t)wave * 16) * IMG_FLOATS;
  float* lds_dst = imgs + wave * 16 * IMG_FLOATS;

#if __has_builtin(__builtin_amdgcn_tensor_load_to_lds)
  {
    uint64_t ga = (uint64_t)(uintptr_t)gsrc;
    uint32_t la = (uint32_t)(uintptr_t)lds_dst;     // low 32 bits = LDS byte offset
    v4u g0;
    g0.x = 1u;                                      // count = 1 valid descriptor
    g0.y = la;                                      // lds_addr [63:32]
    g0.z = (uint32_t)ga;                            // global_addr [95:64]
    g0.w = ((uint32_t)(ga >> 32) & 0x01FFFFFFu) | 0x80000000u; // addr[56:32], type=2
    v8i g1;
    g1[0] = 0x00030000;                             // data_size = 3 (8-byte units)
    g1[1] = (int)(1024u << 16);                     // tensor_dim0 = 1024 (low 16 @ [63:48])
    g1[2] = (int)(1u << 16);                        // tensor_dim1 = 1
    g1[3] = (int)(1024u << 16);                     // tile_dim0 = 1024 @ [127:112]
    g1[4] = 0;                                      // tile_dim1/2 = 0 (1-D tile)
    g1[5] = 1024;                                   // tensor_dim0_stride = 1024
    g1[6] = 0;
    g1[7] = 0;
    v4i gz4 = {0, 0, 0, 0};
    v8i gz8 = {0, 0, 0, 0, 0, 0, 0, 0};
    // clang-23 / therock lane: 6-arg form (extra 8-dword group; zero = unused)
    __builtin_amdgcn_tensor_load_to_lds(g0, g1, gz4, gz4, gz8, 0);
  }
#define USED_TDM 1
#else
  // fallback: cooperative B128 copy global -> LDS
  for (int i = lane; i < 16 * IMG_FLOATS / 4; i += 32)
    ((float4*)lds_dst)[i] = ((const float4*)gsrc)[i];
#define USED_TDM 0
#endif

  // ---------- build Kexp tiles in LDS, exact WMMA A-operand layout ----------
  // A 16x4 f32 layout: lane l holds row M=l%16; VGPR v holds K = v + 2*(l/16).
  // Stored so ds_load_b64 at (tile*64 + lane*2) yields {K=2h, K=2h+1}.
  for (int idx = tid; idx < KEXP_FLOATS; idx += 256) {
    int tt  = idx >> 6;
    int rem = idx & 63;
    int l   = rem >> 1;
    int v   = rem & 1;
    int t = 0;
#pragma unroll
    for (int u = 1; u < 6; ++u) if (tt >= tbase[u]) t = u;
    int ks = kslo[t] + (tt - tbase[t]);
    int p  = 16 * t + (l & 15);                 // output slot (row r=p/6, col c=p%6)
    int q  = 4 * ks + 2 * (l >> 4) + v;         // input pixel (row q/8, col q%8)
    float val = 0.f;
    if (p < 84) {
      int r = p / 6, c = p - 6 * r;
      int i = (q >> 3) - r;
      int j = (q & 7) - c;
      if ((unsigned)i < 3u && (unsigned)j < 3u) val = kern[i * 3 + j];
    }
    kexp[idx] = val;
  }
  __syncthreads();                               // kexp ready for all waves
#if USED_TDM
  __builtin_amdgcn_s_wait_tensorcnt(0);          // this wave's images landed in LDS
#endif

  // ---------- per-wave GEMM: D[96x16] = Kexp[96x128] * X[128x16] ----------
  const float* aB = kexp + lane * 2;                              // + tile*64
  const float* bB = lds_dst + ln * IMG_FLOATS + h * 2;            // + ks*4
  float* obase = out + (imgBlock + wave * 16 + ln) * IMG_FLOATS + h * 8;

#pragma unroll
  for (int t = 0; t < 6; ++t) {
    v8f acc = {0.f, 0.f, 0.f, 0.f, 0.f, 0.f, 0.f, 0.f};
#pragma unroll
    for (int ks = kslo[t]; ks < kshi[t]; ++ks) {
      v2f a = *(const v2f*)(aB + (tbase[t] + ks - kslo[t]) * 64); // ds_load_b64
      v2f b = *(const v2f*)(bB + ks * 4);                         // ds_load_b64
      acc = __builtin_amdgcn_wmma_f32_16x16x4_f32(
          false, a, false, b, (short)0, acc, false, false);
    }
    // D layout: VGPR v, half h -> M = v + 8h; lane ln -> image n.
    // lane stores p = 16t + 8h + {0..7} for its image, fused ReLU.
    float4 r0 = make_float4(fmaxf(acc[0], 0.f), fmaxf(acc[1], 0.f),
                            fmaxf(acc[2], 0.f), fmaxf(acc[3], 0.f));
    float4 r1 = make_float4(fmaxf(acc[4], 0.f), fmaxf(acc[5], 0.f),
                            fmaxf(acc[6], 0.f), fmaxf(acc[7], 0.f));
    *(float4*)(obase + 16 * t)     = r0;          // global_store_b128
    *(float4*)(obase + 16 * t + 4) = r1;
  }
  // slots 84..95 are exact zeros from Kexp's zero rows; zero-fill 96..127
  const float4 z4 = make_float4(0.f, 0.f, 0.f, 0.f);
  *(float4*)(obase + 96)  = z4;
  *(float4*)(obase + 100) = z4;
  *(float4*)(obase + 112) = z4;
  *(float4*)(obase + 116) = z4;
}

extern "C" void kernel_launch(void* const* d_in, const int* in_sizes, int n_in,
                              void* d_out, int out_size, void* d_ws, size_t ws_size,
                              hipStream_t stream) {
  const float* x    = (const float*)d_in[0];   // (256,1024,128) f32
  const float* kern = (const float*)d_in[1];   // (3,3) f32
  float* out = (float*)d_out;                  // (256,1024,128) f32

  int images = in_sizes[0] / IMG_FLOATS;       // 262144
  int blocks = images / IMGS_PER_WG;           // 2048
  conv_relu_wmma<<<dim3(blocks), dim3(256), SMEM_BYTES, stream>>>(x, kern, out);
}